// Kinematics_for_yumi_29927332119227
// MI455X (gfx1250) — compile-verified
//
#include <hip/hip_runtime.h>
#include <cstdint>
#include <cstddef>

#ifndef __has_builtin
#define __has_builtin(x) 0
#endif

#define FK_B      1024
#define FK_T      256
#define FK_NJ     14
#define FK_NELEM  (FK_B * FK_T)
#define FK_BLOCK  256

// ---------------------------------------------------------------------------
// CDNA5 feature probes (compile can never fail; stderr reports selection).
// ---------------------------------------------------------------------------
#if __has_builtin(__builtin_amdgcn_global_load_async_to_lds_b64)
#define FK_ASYNC_LDS 1
#pragma message("FK gfx1250: USING __builtin_amdgcn_global_load_async_to_lds_b64")
#else
#define FK_ASYNC_LDS 0
#pragma message("FK gfx1250: async-to-LDS builtin NOT available -> plain LDS staging")
#endif

#if __has_builtin(__builtin_amdgcn_s_wait_asynccnt)
#define FK_WAIT_ASYNC() __builtin_amdgcn_s_wait_asynccnt(0)
#else
#define FK_WAIT_ASYNC() asm volatile("s_wait_asynccnt 0" ::: "memory")
#endif

#if __has_builtin(__builtin_amdgcn_wmma_f32_16x16x4_f32)
#define FK_WMMA 1
#pragma message("FK gfx1250: USING __builtin_amdgcn_wmma_f32_16x16x4_f32 for STATIC precompute")
#else
#define FK_WMMA 0
#pragma message("FK gfx1250: wmma f32 16x16x4 builtin NOT available -> analytic precompute")
#endif

// ---------------------------------------------------------------------------
// Constants exactly as in the reference source (float64).
// ---------------------------------------------------------------------------
__constant__ double c_TL[7][3] = {
    { 0.05355, -0.0725,  0.41492},
    { 0.03,     0.0,     0.1    },
    {-0.03,     0.17283, 0.0    },
    {-0.04188,  0.0,     0.07873},
    { 0.0405,   0.16461, 0.0    },
    {-0.027,    0.0,     0.10039},
    { 0.027,    0.029,   0.0    }};

__constant__ double c_RL[7][3] = {
    {-0.9795,        -0.5682,        -2.3155       },
    { 1.57079632679,  0.0,            0.0          },
    {-1.57079632679,  0.0,            0.0          },
    { 1.57079632679, -1.57079632679,  0.0          },
    {-1.57079632679,  0.0,            0.0          },
    { 1.57079632679,  0.0,            0.0          },
    {-1.57079632679,  0.0,            0.0          }};

// ---------------------------------------------------------------------------
// Kernel 1: STATIC[j] rotation = Rz(y) @ Ry(p) @ Rx(r); translation = t_j.
// Stored column-major per joint: col0[3], col1[3], col2[3], trans[3].
// Right-arm joints (7..13): t.y negated, roll & yaw negated.
//
// The two 4x4 matrix products per joint are computed on the WMMA engine
// (V_WMMA_F32_16X16X4_F32), batching 4 joints per 16x16 tile block-diagonally:
//   A = stack[Rz_i] (16x4), B = [Ry_0|Ry_1|Ry_2|Ry_3] (4x16),
//   C block(i,i) = Rz_i @ Ry_i ; then again with B' = Rx blocks.
// Diagonal-block re-slotting goes through LDS using the documented layouts:
//   A[m][k] -> VGPR k&1, lane m + 16*(k>>1)
//   C[m][n] -> VGPR m&7, lane n + 16*(m>>3)
// Runs once on a single wave; performance-irrelevant, numerically f32.
// ---------------------------------------------------------------------------
__global__ __launch_bounds__(32) void fk_precompute_static(float* __restrict__ S)
{
    __shared__ float zc[16], zs[16], yc[16], ys[16], xc[16], xs[16];
#if FK_WMMA
    __shared__ float Cbuf[16][16];
#endif

    const int l = threadIdx.x;

    // ---- Phase 0: per-joint trig in float64 (as the reference), cast to f32.
    if (l < 16) {
        float czv = 1.f, szv = 0.f, cyv = 1.f, syv = 0.f, cxv = 1.f, sxv = 0.f;
        if (l < FK_NJ) {
            int    base = (l < 7) ? l : l - 7;
            double sgn  = (l < 7) ? 1.0 : -1.0;
            double r = c_RL[base][0] * sgn;
            double p = c_RL[base][1];
            double y = c_RL[base][2] * sgn;
            cxv = (float)cos(r); sxv = (float)sin(r);
            cyv = (float)cos(p); syv = (float)sin(p);
            czv = (float)cos(y); szv = (float)sin(y);
            float* o = S + l * 12;
            o[9]  = (float)c_TL[base][0];
            o[10] = (float)(c_TL[base][1] * sgn);
            o[11] = (float)c_TL[base][2];
        }
        zc[l] = czv; zs[l] = szv;
        yc[l] = cyv; ys[l] = syv;
        xc[l] = cxv; xs[l] = sxv;
    }
    __syncthreads();

    // 4x4 homogeneous elemental rotations (row r, col c) for joint j.
    auto rzE = [&](int j, int r, int c) -> float {
        if (r == 0) return c == 0 ? zc[j] : (c == 1 ? -zs[j] : 0.f);
        if (r == 1) return c == 0 ? zs[j] : (c == 1 ?  zc[j] : 0.f);
        return (r == c) ? 1.f : 0.f;
    };
    auto ryE = [&](int j, int r, int c) -> float {
        if (r == 0) return c == 0 ?  yc[j] : (c == 2 ? ys[j] : 0.f);
        if (r == 2) return c == 0 ? -ys[j] : (c == 2 ? yc[j] : 0.f);
        return (r == c) ? 1.f : 0.f;
    };
    auto rxE = [&](int j, int r, int c) -> float {
        if (r == 1) return c == 1 ? xc[j] : (c == 2 ? -xs[j] : 0.f);
        if (r == 2) return c == 1 ? xs[j] : (c == 2 ?  xc[j] : 0.f);
        return (r == c) ? 1.f : 0.f;
    };

#if FK_WMMA
    typedef float v2f __attribute__((ext_vector_type(2)));
    typedef float v8f __attribute__((ext_vector_type(8)));

    // Per-lane coordinates in the A / B operand layouts (wave32).
    const int mrow = (l < 16) ? l : l - 16;   // A: matrix row
    const int kk   = (l < 16) ? 0 : 2;        // A: first K slot in this half
    const int ncol = l & 15;                  // B: column
    const int kB   = (l < 16) ? 0 : 1;        // B: row held in VGPR0 by this lane
    const int jA   = (mrow >> 2);             // joint-in-group from A row
    const int rA   = mrow & 3;
    const int jBo  = (ncol >> 2);             // joint-in-group from B col
    const int cB   = ncol & 3;

    for (int g = 0; g < 4; ++g) {             // 4 groups x 4 joints = 16 slots
        const int ja = g * 4 + jA;
        const int jb = g * 4 + jBo;

        v2f a, b;
        a.x = rzE(ja, rA, kk);
        a.y = rzE(ja, rA, kk + 1);
        b.x = ryE(jb, kB,     cB);
        b.y = ryE(jb, kB + 2, cB);

        v8f cz = {};
        v8f C1 = __builtin_amdgcn_wmma_f32_16x16x4_f32(
                     false, a, false, b, (short)0, cz, false, false);

        __syncthreads();                      // Cbuf reused across groups
        {                                     // scatter C1 (layout: vgpr m&7, lane n+16*(m>>3))
            const int mb = (l >= 16) ? 8 : 0;
#pragma unroll
            for (int v = 0; v < 8; ++v) Cbuf[v + mb][ncol] = C1[v];
        }
        __syncthreads();

        // A' = diagonal blocks of C1 (= Rz@Ry per joint), B' = Rx blocks.
        v2f a2, b2;
        a2.x = Cbuf[mrow][((mrow >> 2) << 2) + kk];
        a2.y = Cbuf[mrow][((mrow >> 2) << 2) + kk + 1];
        b2.x = rxE(jb, kB,     cB);
        b2.y = rxE(jb, kB + 2, cB);

        v8f C2 = __builtin_amdgcn_wmma_f32_16x16x4_f32(
                     false, a2, false, b2, (short)0, cz, false, false);

        // Diagonal 3x3 blocks -> S (column-major per joint).
        {
            const int mb = (l >= 16) ? 8 : 0;
#pragma unroll
            for (int v = 0; v < 8; ++v) {
                const int m = v + mb;
                if ((m >> 2) == (ncol >> 2)) {
                    const int j = g * 4 + (m >> 2);
                    const int r = m & 3, c = ncol & 3;
                    if (j < FK_NJ && r < 3 && c < 3)
                        S[j * 12 + c * 3 + r] = C2[v];
                }
            }
        }
    }
#else
    // Analytic fallback: R = Rz(y)Ry(p)Rx(r) closed form.
    if (l < FK_NJ) {
        float cr = xc[l], sr = xs[l], cp = yc[l], sp = ys[l], cy = zc[l], sy = zs[l];
        float* o = S + l * 12;
        o[0] = cy*cp;            o[1] = sy*cp;            o[2] = -sp;      // col 0
        o[3] = cy*sp*sr - sy*cr; o[4] = sy*sp*sr + cy*cr; o[5] = cp*sr;    // col 1
        o[6] = cy*sp*cr + sy*sr; o[7] = sy*sp*cr - cy*sr; o[8] = cp*cr;    // col 2
    }
#endif
}

// ---------------------------------------------------------------------------
// Kernel 2: one thread per (b,t) element. Memory-bound (~44 MB @ 23.3 TB/s).
// ---------------------------------------------------------------------------
__global__ __launch_bounds__(FK_BLOCK) void fk_kernel(
    const float* __restrict__ eulers,   // (B,T,14)
    const float* __restrict__ Sg,       // 14 x 12 precomputed STATIC (col-major)
    float* __restrict__ Qout,           // (B,T,4,4)
    float* __restrict__ Pout)           // (B,T,4,3)
{
    __shared__ float sang[FK_BLOCK * FK_NJ];

    const int tid = threadIdx.x;
    const int idx = blockIdx.x * FK_BLOCK + tid;
    if (idx >= FK_NELEM) return;

    const float* gsrc   = eulers + (size_t)idx * FK_NJ;   // 56 B, 8-byte aligned
    float*       lslice = &sang[tid * FK_NJ];             // stride 14 words: conflict-free (64 banks, wave32)

#if FK_ASYNC_LDS
    {
        typedef int v2i __attribute__((vector_size(2 * sizeof(int))));
        typedef __attribute__((address_space(1))) v2i GV2;
        typedef __attribute__((address_space(3))) v2i LV2;
        GV2* g = (GV2*)gsrc;
        LV2* l = (LV2*)lslice;
#pragma unroll
        for (int i = 0; i < 7; ++i)    // 7 x 8B = this thread's 14 angles
            __builtin_amdgcn_global_load_async_to_lds_b64(g + i, l + i, 0, 0);
        FK_WAIT_ASYNC();               // own-slice copy: no cross-wave barrier needed
        asm volatile("" ::: "memory");
    }
#else
#pragma unroll
    for (int k = 0; k < FK_NJ; ++k) lslice[k] = gsrc[k];
#endif

    float q[16];
    float p[12];

    // Chain state: row-major R, translation t.
    float R00, R01, R02, R10, R11, R12, R20, R21, R22, t0, t1, t2;

    auto emit = [&](float a00, float a01, float a02,
                    float a10, float a11, float a12,
                    float a20, float a21, float a22,
                    float b0, float b1, float b2, int slot) {
        float tr  = a00 + a11 + a22;
        float w   = 0.5f * sqrtf(fmaxf(1.0f + tr, 1e-8f));
        float d   = fmaxf(4.0f * w, 1e-6f);
        float inv = 1.0f / d;
        q[slot * 4 + 0] = w;
        q[slot * 4 + 1] = (a21 - a12) * inv;
        q[slot * 4 + 2] = (a02 - a20) * inv;
        q[slot * 4 + 3] = (a10 - a01) * inv;
        p[slot * 3 + 0] = b0;
        p[slot * 3 + 1] = b1;
        p[slot * 3 + 2] = b2;
    };

    auto run_arm = [&](int joff, int eslot, int wslot) {
        // ROOT_T = Txyz(-ORIGIN), ORIGIN = (0.05355, 0, 0.41492)
        R00 = 1.f; R01 = 0.f; R02 = 0.f;
        R10 = 0.f; R11 = 1.f; R12 = 0.f;
        R20 = 0.f; R21 = 0.f; R22 = 1.f;
        t0 = -0.05355f; t1 = 0.0f; t2 = -0.41492f;

#pragma unroll
        for (int k = 0; k < 7; ++k) {
            const float* Sj = Sg + (size_t)(joff + k) * 12;   // uniform -> scalar loads
            float s00 = Sj[0],  s10 = Sj[1],  s20 = Sj[2];
            float s01 = Sj[3],  s11 = Sj[4],  s21 = Sj[5];
            float s02 = Sj[6],  s12 = Sj[7],  s22 = Sj[8];
            float st0 = Sj[9],  st1 = Sj[10], st2 = Sj[11];

            float th = lslice[joff + k];
            float sn, cs;
            __sincosf(th, &sn, &cs);

            // U = T @ S  (columns u, v, w ; translation nt)
            float u0 = R00*s00 + R01*s10 + R02*s20;
            float u1 = R10*s00 + R11*s10 + R12*s20;
            float u2 = R20*s00 + R21*s10 + R22*s20;
            float v0 = R00*s01 + R01*s11 + R02*s21;
            float v1 = R10*s01 + R11*s11 + R12*s21;
            float v2 = R20*s01 + R21*s11 + R22*s21;
            float w0 = R00*s02 + R01*s12 + R02*s22;
            float w1 = R10*s02 + R11*s12 + R12*s22;
            float w2 = R20*s02 + R21*s12 + R22*s22;
            float nt0 = R00*st0 + R01*st1 + R02*st2 + t0;
            float nt1 = R10*st0 + R11*st1 + R12*st2 + t1;
            float nt2 = R20*st0 + R21*st1 + R22*st2 + t2;

            // T' = U @ Rz(th): col0' = c*u + s*v ; col1' = c*v - s*u ; col2' = w
            R00 = cs*u0 + sn*v0;  R01 = cs*v0 - sn*u0;  R02 = w0;
            R10 = cs*u1 + sn*v1;  R11 = cs*v1 - sn*u1;  R12 = w1;
            R20 = cs*u2 + sn*v2;  R21 = cs*v2 - sn*u2;  R22 = w2;
            t0 = nt0; t1 = nt1; t2 = nt2;

            if (k == 3)   // ELBOW_K
                emit(R00, R01, R02, R10, R11, R12, R20, R21, R22, t0, t1, t2, eslot);
        }
        emit(R00, R01, R02, R10, R11, R12, R20, R21, R22, t0, t1, t2, wslot);
    };

    run_arm(0, 0, 1);   // left arm  -> slots eL=0, wL=1
    run_arm(7, 2, 3);   // right arm -> slots eR=2, wR=3

    // Vectorized stores: 64 B of Q + 48 B of P per thread, fully coalesced.
    float4* Q4 = reinterpret_cast<float4*>(Qout) + (size_t)idx * 4;
    Q4[0] = make_float4(q[0],  q[1],  q[2],  q[3]);
    Q4[1] = make_float4(q[4],  q[5],  q[6],  q[7]);
    Q4[2] = make_float4(q[8],  q[9],  q[10], q[11]);
    Q4[3] = make_float4(q[12], q[13], q[14], q[15]);

    float4* P4 = reinterpret_cast<float4*>(Pout) + (size_t)idx * 3;
    P4[0] = make_float4(p[0], p[1], p[2],  p[3]);
    P4[1] = make_float4(p[4], p[5], p[6],  p[7]);
    P4[2] = make_float4(p[8], p[9], p[10], p[11]);
}

// ---------------------------------------------------------------------------
// Host launcher
// ---------------------------------------------------------------------------
extern "C" void kernel_launch(void* const* d_in, const int* in_sizes, int n_in,
                              void* d_out, int out_size, void* d_ws, size_t ws_size,
                              hipStream_t stream)
{
    (void)in_sizes; (void)n_in; (void)out_size; (void)ws_size;

    const float* eulers = (const float*)d_in[0];
    float*       Sg     = (float*)d_ws;                       // 14*12 floats = 672 B scratch
    float*       Qout   = (float*)d_out;                      // (B,T,4,4) = 4,194,304 floats
    float*       Pout   = Qout + (size_t)FK_NELEM * 16;       // (B,T,4,3) = 3,145,728 floats

    fk_precompute_static<<<1, 32, 0, stream>>>(Sg);
    fk_kernel<<<FK_NELEM / FK_BLOCK, FK_BLOCK, 0, stream>>>(eulers, Sg, Qout, Pout);
}